// QNNClassifier_75780402971315
// MI455X (gfx1250) — compile-verified
//
#include <hip/hip_runtime.h>

typedef float v2f __attribute__((ext_vector_type(2)));
typedef float v8f __attribute__((ext_vector_type(8)));

#define NQ      7
#define DIM     128       // 2^NQ
#define NL      4
#define BATCH   32768
#define NCLS    22
#define BT      32        // sim threads per block (one wave32)

// ---------------------------------------------------------------------------
// CDNA5 f32 WMMA: D(16x16,f32) = A(16x4,f32) * B(4x16,f32) + C
// A/B fragments: lane l, reg j -> {M|N}=l&15, K = 2*(l>>4)+j
// C/D fragment : lane l, reg r -> N=l&15, M = r + 8*(l>>4)
// ---------------------------------------------------------------------------
static __device__ __forceinline__ v8f wmma4(v2f a, v2f b, v8f c) {
  return __builtin_amdgcn_wmma_f32_16x16x4_f32(
      /*neg_a=*/false, a, /*neg_b=*/false, b,
      /*c_mod=*/(short)0, c, /*reuse_a=*/false, /*reuse_b=*/false);
}

struct C2x2 { float r00,i00,r01,i01,r10,i10,r11,i11; };

// ---------------------------------------------------------------------------
// Kernel 0: precompute the 28 Rot gate matrices (batch-independent) once,
// instead of 32768 threads each doing 28 x 6 transcendentals redundantly.
// gbuf[(l*NQ+w)*8 + {0..7}] = {r00,i00,r01,i01,r10,i10,r11,i11}
// ---------------------------------------------------------------------------
__global__ __launch_bounds__(32) void prep_kernel(const float* __restrict__ qw,
                                                  float* __restrict__ gbuf) {
  const int t = threadIdx.x;
  if (t < NL * NQ) {
    const float phi = qw[t * 3 + 0];
    const float th  = qw[t * 3 + 1];
    const float om  = qw[t * 3 + 2];
    const float c  = cosf(0.5f * th), s  = sinf(0.5f * th);
    const float aa = 0.5f * (phi + om), bb = 0.5f * (phi - om);
    const float ca = cosf(aa), sa = sinf(aa);
    const float cb = cosf(bb), sb = sinf(bb);
    // m00 = c*e^{-i a}, m01 = -s*e^{i b}, m10 = s*e^{-i b}, m11 = c*e^{i a}
    float* g = gbuf + t * 8;
    g[0] =  c * ca;  g[1] = -c * sa;
    g[2] = -s * cb;  g[3] = -s * sb;
    g[4] =  s * cb;  g[5] = -s * sb;
    g[6] =  c * ca;  g[7] =  c * sa;
  }
}

static __device__ __forceinline__ C2x2 load_gate(const float* __restrict__ g,
                                                 int idx) {
  // uniform address -> scalar-cached loads
  C2x2 U;
  U.r00 = g[idx*8+0]; U.i00 = g[idx*8+1];
  U.r01 = g[idx*8+2]; U.i01 = g[idx*8+3];
  U.r10 = g[idx*8+4]; U.i10 = g[idx*8+5];
  U.r11 = g[idx*8+6]; U.i11 = g[idx*8+7];
  return U;
}

static __device__ __forceinline__ int insert0(int v, int pos) {
  const int low = (1 << pos) - 1;
  return ((v & ~low) << 1) | (v & low);
}

// Single-qubit gate on logical qubit with physical XOR-mask m, parity row r.
static __device__ __forceinline__ void apply1(float* sre, float* sig, int tid,
                                              const C2x2& U, int m, int r) {
  const int bsel = __builtin_ctz(m);
#pragma unroll 4
  for (int p = 0; p < DIM / 2; ++p) {
    const int j0  = insert0(p, bsel);
    const int par = __popc(j0 & r) & 1;       // logical bit of j0
    const int ia  = par ? (j0 ^ m) : j0;      // logical-0 member
    const int ib  = ia ^ m;                   // logical-1 member
    const float r0 = sre[ia * BT + tid], g0 = sig[ia * BT + tid];
    const float r1 = sre[ib * BT + tid], g1 = sig[ib * BT + tid];
    const float nr0 = U.r00*r0 - U.i00*g0 + U.r01*r1 - U.i01*g1;
    const float ng0 = U.r00*g0 + U.i00*r0 + U.r01*g1 + U.i01*r1;
    const float nr1 = U.r10*r0 - U.i10*g0 + U.r11*r1 - U.i11*g1;
    const float ng1 = U.r10*g0 + U.i10*r0 + U.r11*g1 + U.i11*r1;
    sre[ia * BT + tid] = nr0;  sig[ia * BT + tid] = ng0;
    sre[ib * BT + tid] = nr1;  sig[ib * BT + tid] = ng1;
  }
}

// Fused two-qubit pass: G = U (x) V acting on logical bits (a,b).
// Each amplitude is read+written once (instead of twice for two 1q passes).
static __device__ __forceinline__ void apply2(float* sre, float* sig, int tid,
                                              const C2x2& U, const C2x2& V,
                                              int m1, int r1, int m2, int r2) {
  const float Ur[2][2] = {{U.r00, U.r01}, {U.r10, U.r11}};
  const float Ui[2][2] = {{U.i00, U.i01}, {U.i10, U.i11}};
  const float Vr[2][2] = {{V.r00, V.r01}, {V.r10, V.r11}};
  const float Vi[2][2] = {{V.i00, V.i01}, {V.i10, V.i11}};
  float Gr[4][4], Gi[4][4];
#pragma unroll
  for (int a = 0; a < 2; ++a)
#pragma unroll
    for (int b = 0; b < 2; ++b)
#pragma unroll
      for (int a2 = 0; a2 < 2; ++a2)
#pragma unroll
        for (int b2 = 0; b2 < 2; ++b2) {
          Gr[a*2+b][a2*2+b2] = Ur[a][a2]*Vr[b][b2] - Ui[a][a2]*Vi[b][b2];
          Gi[a*2+b][a2*2+b2] = Ur[a][a2]*Vi[b][b2] + Ui[a][a2]*Vr[b][b2];
        }
  // Pick clear-bit positions so the projection of {m1,m2} on (b1,b2) is an
  // invertible GF(2) 2x2 -> the 32 representatives tile all 128 amps.
  int b1, b2;
  if (m1 & ~m2) { b1 = __builtin_ctz(m1 & ~m2); b2 = __builtin_ctz(m2); }
  else          { b1 = __builtin_ctz(m1);       b2 = __builtin_ctz(m2 & ~m1); }
  const int lo = (b1 < b2) ? b1 : b2;
  const int hi = (b1 < b2) ? b2 : b1;
#pragma unroll 4
  for (int p = 0; p < DIM / 4; ++p) {
    const int j  = insert0(insert0(p, lo), hi);
    const int p1 = __popc(j & r1) & 1;
    const int p2 = __popc(j & r2) & 1;
    const int j00 = j ^ (p1 ? m1 : 0) ^ (p2 ? m2 : 0);  // logical (0,0)
    int idx[4];
    idx[0] = j00;            // (a,b) = (0,0)
    idx[1] = j00 ^ m2;       // (0,1)
    idx[2] = j00 ^ m1;       // (1,0)
    idx[3] = j00 ^ m1 ^ m2;  // (1,1)
    float ar[4], ai[4];
#pragma unroll
    for (int e = 0; e < 4; ++e) {
      ar[e] = sre[idx[e] * BT + tid];
      ai[e] = sig[idx[e] * BT + tid];
    }
#pragma unroll
    for (int o = 0; o < 4; ++o) {
      float orr = 0.f, oii = 0.f;
#pragma unroll
      for (int e = 0; e < 4; ++e) {
        orr = fmaf(Gr[o][e], ar[e], orr);
        orr = fmaf(-Gi[o][e], ai[e], orr);
        oii = fmaf(Gr[o][e], ai[e], oii);
        oii = fmaf(Gi[o][e], ar[e], oii);
      }
      sre[idx[o] * BT + tid] = orr;
      sig[idx[o] * BT + tid] = oii;
    }
  }
}

// ---------------------------------------------------------------------------
// Kernel 1: fused  h = x@Wp.T + bp  ->  7-qubit statevector sim  ->  q (B,7)
// One thread = one batch element; state in LDS interleaved s[idx*BT+tid]
// (32 lanes -> 32 consecutive banks, conflict-free).  CNOT permutations are
// linear maps over GF(2)^7 and are never executed: we track the cumulative
// relabeling sigma via XOR-masks m[w] (columns of sigma) and parity rows
// r[w] (rows of sigma^-1).  CNOT(c,t): m[c]^=m[t]; r[t]^=r[c].
// All layer/qubit loops fully unrolled -> masks fold to constants.
// ---------------------------------------------------------------------------
__global__ __launch_bounds__(BT) void qsim_kernel(
    const float* __restrict__ x,  const float* __restrict__ Wp,
    const float* __restrict__ bp, const float* __restrict__ gbuf,
    float* __restrict__ qout) {
  __shared__ float sre[DIM * BT];
  __shared__ float sig[DIM * BT];
  const int tid = threadIdx.x;
  const int b   = blockIdx.x * BT + tid;

  // h = x @ Wp.T + bp   (Wp/bp uniform -> scalar-cached loads)
  float xr[NQ];
#pragma unroll
  for (int j = 0; j < NQ; ++j) xr[j] = x[b * NQ + j];
  float cw_[NQ], sw_[NQ];
#pragma unroll
  for (int t = 0; t < NQ; ++t) {
    float acc = bp[t];
#pragma unroll
    for (int j = 0; j < NQ; ++j) acc = fmaf(xr[j], Wp[t * NQ + j], acc);
    const float half = 0.5f * acc;
    cw_[t] = cosf(half);
    sw_[t] = sinf(half);
  }

  // RY layer on |0..0> is a product state: amp(i) = prod_w (bit_w ? s_w : c_w)
  for (int i = 0; i < DIM; ++i) {
    float v = 1.f;
#pragma unroll
    for (int w = 0; w < NQ; ++w) v *= ((i >> (NQ - 1 - w)) & 1) ? sw_[w] : cw_[w];
    sre[i * BT + tid] = v;
    sig[i * BT + tid] = 0.f;
  }

  // GF(2) relabeling state: sigma = identity initially.
  int m[NQ], r[NQ];
#pragma unroll
  for (int w = 0; w < NQ; ++w) { m[w] = 1 << (NQ-1-w); r[w] = 1 << (NQ-1-w); }

#pragma unroll
  for (int l = 0; l < NL; ++l) {
    // Rot gates: fuse (0,1),(2,3),(4,5) into 4x4 passes; qubit 6 single.
#pragma unroll
    for (int g = 0; g < 3; ++g) {
      const C2x2 U = load_gate(gbuf, l * NQ + 2*g);
      const C2x2 V = load_gate(gbuf, l * NQ + 2*g + 1);
      apply2(sre, sig, tid, U, V, m[2*g], r[2*g], m[2*g+1], r[2*g+1]);
    }
    {
      const C2x2 U = load_gate(gbuf, l * NQ + 6);
      apply1(sre, sig, tid, U, m[6], r[6]);
    }
    // CNOT ring: pure index relabeling, no data movement.
#pragma unroll
    for (int cq = 0; cq < NQ; ++cq) {
      const int tq = (cq + l + 1) % NQ;
      m[cq] ^= m[tq];
      r[tq] ^= r[cq];
    }
  }

  // q[k] = sum_j |amp_j|^2 * (1 - 2*parity(j & r[k]))
  float q[NQ];
#pragma unroll
  for (int k = 0; k < NQ; ++k) q[k] = 0.f;
  for (int j = 0; j < DIM; ++j) {
    const float rr = sre[j * BT + tid], gg = sig[j * BT + tid];
    const float pr = rr * rr + gg * gg;
#pragma unroll
    for (int k = 0; k < NQ; ++k) q[k] += (__popc(j & r[k]) & 1) ? -pr : pr;
  }
#pragma unroll
  for (int k = 0; k < NQ; ++k) qout[b * 8 + k] = q[k];
  qout[b * 8 + 7] = 0.f;  // K-pad so the WMMA A-fragment loads need no guard
}

// ---------------------------------------------------------------------------
// Kernel 2: MLP head via V_WMMA_F32_16X16X4_F32.
// One wave per 16-row batch tile; 4 waves (4 tiles) per block.
//   GEMM1: (16x8) @ (8x16)   -> 2 WMMA  (q @ W1.T, K padded 7->8)
//   GEMM2: (16x16) @ (16x32) -> 8 WMMA  (h1 @ W2.T, N padded 22->32)
// h1 is restaged through LDS to convert C-fragment layout -> A-fragment.
// ---------------------------------------------------------------------------
__global__ __launch_bounds__(128) void head_kernel(
    const float* __restrict__ qbuf, const float* __restrict__ W1,
    const float* __restrict__ b1,   const float* __restrict__ W2,
    const float* __restrict__ b2,   float* __restrict__ out) {
  __shared__ float stage[4][16 * 16];
  const int tid  = threadIdx.x;
  const int lane = tid & 31;
  const int wave = tid >> 5;
  const int ln16 = lane & 15;
  const int kx   = lane >> 4;                 // 0 or 1 (lane half)
  const int m0   = (blockIdx.x * 4 + wave) * 16;
  float* st = stage[wave];

  // ---- GEMM1: q(16x8) @ W1.T(8x16) ----
  v8f c1 = {0.f, 0.f, 0.f, 0.f, 0.f, 0.f, 0.f, 0.f};
#pragma unroll
  for (int kb = 0; kb < 2; ++kb) {
    const int k0 = kb * 4 + 2 * kx;
    v2f a, bm;
    a.x  = qbuf[(m0 + ln16) * 8 + k0];
    a.y  = qbuf[(m0 + ln16) * 8 + k0 + 1];
    bm.x = (k0     < NQ) ? W1[ln16 * NQ + k0]     : 0.f;
    bm.y = (k0 + 1 < NQ) ? W1[ln16 * NQ + k0 + 1] : 0.f;
    c1 = wmma4(a, bm, c1);
  }
  // bias + ReLU in C-layout, restage as row-major 16x16 tile
  const float bias1 = b1[ln16];
#pragma unroll
  for (int rr = 0; rr < 8; ++rr) {
    float v = c1[rr] + bias1;
    v = fmaxf(v, 0.f);
    st[(rr + 8 * kx) * 16 + ln16] = v;   // st[m][n] = h1[m][n]
  }
  __syncthreads();

  // ---- GEMM2: h1(16x16) @ W2.T(16x32-pad) ----
  v8f d0 = {0.f, 0.f, 0.f, 0.f, 0.f, 0.f, 0.f, 0.f};
  v8f d1 = {0.f, 0.f, 0.f, 0.f, 0.f, 0.f, 0.f, 0.f};
#pragma unroll
  for (int kb = 0; kb < 4; ++kb) {
    const int k0 = kb * 4 + 2 * kx;
    v2f a2, bA, bB;
    a2.x = st[ln16 * 16 + k0];
    a2.y = st[ln16 * 16 + k0 + 1];
    bA.x = W2[ln16 * 16 + k0];            // N-tile 0: n = ln16 < 16 (< 22)
    bA.y = W2[ln16 * 16 + k0 + 1];
    const int n1 = 16 + ln16;             // N-tile 1: n = 16..31, valid < 22
    bB.x = (n1 < NCLS) ? W2[n1 * 16 + k0]     : 0.f;
    bB.y = (n1 < NCLS) ? W2[n1 * 16 + k0 + 1] : 0.f;
    d0 = wmma4(a2, bA, d0);
    d1 = wmma4(a2, bB, d1);
  }

  // ---- epilogue: + b2, store (B,22) ----
#pragma unroll
  for (int rr = 0; rr < 8; ++rr) {
    const int M = m0 + rr + 8 * kx;
    out[M * NCLS + ln16] = d0[rr] + b2[ln16];
    const int n1 = 16 + ln16;
    if (n1 < NCLS) out[M * NCLS + n1] = d1[rr] + b2[n1];
  }
}

// ---------------------------------------------------------------------------
extern "C" void kernel_launch(void* const* d_in, const int* in_sizes, int n_in,
                              void* d_out, int out_size, void* d_ws, size_t ws_size,
                              hipStream_t stream) {
  const float* x   = (const float*)d_in[0];
  const float* Wp  = (const float*)d_in[1];
  const float* bp  = (const float*)d_in[2];
  const float* qw  = (const float*)d_in[3];
  const float* W1  = (const float*)d_in[4];
  const float* b1  = (const float*)d_in[5];
  const float* W2  = (const float*)d_in[6];
  const float* b2  = (const float*)d_in[7];
  float* out  = (float*)d_out;
  float* gbuf = (float*)d_ws;            // 28*8 floats (1 KB region)
  float* qbuf = gbuf + 256;              // BATCH * 8 floats = 1 MB

  prep_kernel<<<1, 32, 0, stream>>>(qw, gbuf);
  qsim_kernel<<<BATCH / BT, BT, 0, stream>>>(x, Wp, bp, gbuf, qbuf);
  head_kernel<<<BATCH / 64, 128, 0, stream>>>(qbuf, W1, b1, W2, b2, out);
}